// SPDLogEuclideanParametrization_20272245637321
// MI455X (gfx1250) — compile-verified
//
#include <hip/hip_runtime.h>

// ---------------------------------------------------------------------------
// expm(symmetric 32x32) via scaling-and-squaring + Paterson-Stockmeyer
// Taylor (degree 15), all matmuls on V_WMMA_F32_16X16X4_F32.
//
// One wave32 owns one matrix. LDS buffers per wave (row stride 36 floats to
// kill bank conflicts on column-strided A-fragment loads while keeping 16B
// alignment):  T, T^2, T^3, T^4, and two ping-pong accumulators.
// ---------------------------------------------------------------------------

typedef float v2f __attribute__((ext_vector_type(2)));
typedef float v8f __attribute__((ext_vector_type(8)));

#define LS   36            // LDS row stride in floats (32 + 4 pad)
#define MSZ  (32 * LS)     // floats per 32x32 LDS matrix buffer

// Taylor coefficients 1/k!
#define C0  1.0f
#define C1  1.0f
#define C2  0.5f
#define C3  1.6666666666666666e-1f
#define C4  4.1666666666666664e-2f
#define C5  8.3333333333333333e-3f
#define C6  1.3888888888888889e-3f
#define C7  1.9841269841269841e-4f
#define C8  2.4801587301587302e-5f
#define C9  2.7557319223985893e-6f
#define C10 2.7557319223985890e-7f
#define C11 2.5052108385441720e-8f
#define C12 2.0876756987868100e-9f
#define C13 1.6059043836821613e-10f
#define C14 1.1470745597729725e-11f
#define C15 7.6471637318198164e-13f

// D = X * Y (+ optional C-init: k0*I + k1*T + k2*P2 + k3*P3), 32x32 fp32.
// 2x2 output tiles of 16x16; K split into 8 slices of 4 for wmma f32 16x16x4.
//
// fp32 WMMA VGPR layouts (CDNA5 ISA 7.12.2):
//   A 16x4 : lane m=lane&15; v[0] holds K = (lane>=16?2:0)+kbase, v[1] = K+1
//   B 4x16 : lane n=lane&15; v[0] holds K = (lane>=16?2:0)+kbase, v[1] = K+1
//   C 16x16: VGPR r -> M = r + (lane>=16?8:0), N = lane&15
template <bool USEC>
__device__ __forceinline__ void mm32(const float* X, const float* Y, float* D,
                                     const float* T, const float* P2,
                                     const float* P3, float k0, float k1,
                                     float k2, float k3, int lane) {
  const int l15 = lane & 15;
  const int hi  = lane >> 4;  // 0 or 1
#pragma unroll
  for (int j = 0; j < 2; ++j) {
    const int n = j * 16 + l15;
    v2f b[8];
#pragma unroll
    for (int kk = 0; kk < 8; ++kk) {
      const int kb = kk * 4 + hi * 2;
      b[kk][0] = Y[(kb + 0) * LS + n];
      b[kk][1] = Y[(kb + 1) * LS + n];
    }
#pragma unroll
    for (int i = 0; i < 2; ++i) {
      v8f acc;
#pragma unroll
      for (int r = 0; r < 8; ++r) {
        const int m = i * 16 + hi * 8 + r;
        float cv = 0.0f;
        if (USEC) {
          const int o = m * LS + n;
          cv = k1 * T[o] + k2 * P2[o] + k3 * P3[o];
          if (m == n) cv += k0;
        }
        acc[r] = cv;
      }
      v2f a[8];
      const float* xr = &X[(i * 16 + l15) * LS];
#pragma unroll
      for (int kk = 0; kk < 8; ++kk) {
        const int kb = kk * 4 + hi * 2;
        a[kk][0] = xr[kb];
        a[kk][1] = xr[kb + 1];
      }
#pragma unroll
      for (int kk = 0; kk < 8; ++kk)
        acc = __builtin_amdgcn_wmma_f32_16x16x4_f32(
            false, a[kk], false, b[kk], (short)0, acc, false, false);
#pragma unroll
      for (int r = 0; r < 8; ++r)
        D[(i * 16 + hi * 8 + r) * LS + n] = acc[r];
    }
  }
}

__global__ void __launch_bounds__(64) expm32_kernel(
    const float* __restrict__ in, float* __restrict__ out, int nmat) {
  __shared__ float lds[2][6][MSZ];
  const int wv   = threadIdx.x >> 5;
  const int lane = threadIdx.x & 31;
  const int mat  = blockIdx.x * 2 + wv;
  if (mat >= nmat) return;  // wave-uniform: EXEC stays all-1s for WMMA

  float* T  = &lds[wv][0][0];
  float* P2 = &lds[wv][1][0];
  float* P3 = &lds[wv][2][0];
  float* P4 = &lds[wv][3][0];
  float* A1 = &lds[wv][4][0];
  float* A2 = &lds[wv][5][0];

  // ---- stage matrix into LDS (coalesced float4 global loads) ----
  const float4* g = (const float4*)(in + (size_t)mat * 1024);
#pragma unroll
  for (int it = 0; it < 8; ++it) {
    const int e   = it * 32 + lane;   // float4 index within matrix
    const float4 v = g[e];
    const int row = e >> 3;           // 8 float4 per row
    const int col = (e & 7) * 4;
    *(float4*)&T[row * LS + col] = v; // 16B aligned: LS and col are mult of 4
  }

  // ---- inf-norm -> scaling exponent s (wave-uniform) ----
  float rs = 0.0f;
  for (int c = 0; c < 32; ++c) rs += fabsf(T[lane * LS + c]);
  for (int off = 16; off > 0; off >>= 1)
    rs = fmaxf(rs, __shfl_xor(rs, off, 32));
  int s = 0;
  while (rs > 1.0f && s < 15) { rs *= 0.5f; ++s; }
  const float sc = ldexpf(1.0f, -s);

  for (int e = lane; e < 1024; e += 32)
    T[(e >> 5) * LS + (e & 31)] *= sc;

  // ---- powers T^2, T^3, T^4 ----
  mm32<false>(T,  T,  P2, T, P2, P3, 0, 0, 0, 0, lane);
  mm32<false>(T,  P2, P3, T, P2, P3, 0, 0, 0, 0, lane);
  mm32<false>(P2, P2, P4, T, P2, P3, 0, 0, 0, 0, lane);

  // ---- A1 = B3 = c12*I + c13*T + c14*T^2 + c15*T^3 (elementwise) ----
  for (int e = lane; e < 1024; e += 32) {
    const int row = e >> 5, col = e & 31, o = row * LS + col;
    float v = C13 * T[o] + C14 * P2[o] + C15 * P3[o];
    if (row == col) v += C12;
    A1[o] = v;
  }

  // ---- Horner in T^4: exp(T) ~= ((B3*P4 + B2)*P4 + B1)*P4 + B0 ----
  mm32<true>(A1, P4, A2, T, P2, P3, C8, C9, C10, C11, lane);
  mm32<true>(A2, P4, A1, T, P2, P3, C4, C5, C6,  C7,  lane);
  mm32<true>(A1, P4, A2, T, P2, P3, C0, C1, C2,  C3,  lane);

  // ---- s squarings: exp(A) = exp(T)^(2^s) ----
  float* cur = A2;
  float* oth = A1;
  for (int q = 0; q < s; ++q) {
    mm32<false>(cur, cur, oth, T, P2, P3, 0, 0, 0, 0, lane);
    float* t = cur; cur = oth; oth = t;
  }

  // ---- write back (coalesced float4 stores) ----
  float4* go = (float4*)(out + (size_t)mat * 1024);
#pragma unroll
  for (int it = 0; it < 8; ++it) {
    const int e   = it * 32 + lane;
    const int row = e >> 3;
    const int col = (e & 7) * 4;
    go[e] = *(const float4*)&cur[row * LS + col];
  }
}

extern "C" void kernel_launch(void* const* d_in, const int* in_sizes, int n_in,
                              void* d_out, int out_size, void* d_ws,
                              size_t ws_size, hipStream_t stream) {
  (void)n_in; (void)out_size; (void)d_ws; (void)ws_size;
  const float* in = (const float*)d_in[0];
  float* out      = (float*)d_out;
  const int nmat  = in_sizes[0] / 1024;       // 65536 matrices of 32x32
  const int blocks = (nmat + 1) / 2;          // 2 waves (2 matrices) / block
  expm32_kernel<<<blocks, 64, 0, stream>>>(in, out, nmat);
}